// DynamicLIF_16741782520531
// MI455X (gfx1250) — compile-verified
//
#include <hip/hip_runtime.h>

// ---------------------------------------------------------------------------
// DynamicLIF scan on MI455X (gfx1250, wave32).
//
// current: [T=200, B=128, N=2048] f32, tau: [N] f32 -> spikes [T,B,N] f32.
//
// Roofline: ~420 MB compulsory HBM traffic (210 MB read + 210 MB write),
// ~5 FLOP per 8 bytes -> purely bandwidth-bound, ~18 us floor @ 23.3 TB/s.
// No matmul anywhere, so WMMA does not apply; the win comes from the CDNA5
// data-movement path: NT-hinted b128 streaming (dataset >> 192 MB L2),
// global_prefetch_b8 along the 1 MB-strided time axis, and keeping the
// membrane state in VGPRs for the entire scan (zero state traffic).
// ---------------------------------------------------------------------------

typedef float f32x4 __attribute__((ext_vector_type(4)));

#define LIF_DT   0.01f
#define LIF_VTH  1.0f

// T is fixed by the reference (divisible by the unroll factor) -> no
// remainder loop, no runtime trip-count SALU logic.
static constexpr int LIF_T = 200;

__launch_bounds__(256, 8)
__global__ void lif_scan_kernel(const float* __restrict__ cur,
                                const float* __restrict__ tau,
                                float* __restrict__ out,
                                int BN, int Nmask) {
    const int tid = blockIdx.x * blockDim.x + threadIdx.x;
    const int idx = tid * 4;                 // flat index into a [B*N] row
    if (idx >= BN) return;

    // Per-thread decay = DT / tau[n], n = idx % N (4 consecutive neurons).
    // One-time cost; full-rate divide is irrelevant outside the scan loop.
    const int n = idx & Nmask;
    const f32x4 tv = *(const f32x4*)(tau + n);
    f32x4 decay;
    decay.x = LIF_DT / tv.x;
    decay.y = LIF_DT / tv.y;
    decay.z = LIF_DT / tv.z;
    decay.w = LIF_DT / tv.w;

    f32x4 v = {0.0f, 0.0f, 0.0f, 0.0f};     // membrane potential (stays in VGPRs)

    const size_t stride = (size_t)BN;        // one time step = B*N floats
    const float* cptr = cur + idx;
    float*       optr = out + idx;

#pragma unroll 4
    for (int s = 0; s < LIF_T; ++s) {
        // Streaming load, non-temporal (each byte read exactly once).
        const f32x4 c = __builtin_nontemporal_load((const f32x4*)cptr);

        // Prefetch 8 steps (8 MB) ahead; speculative prefetches past the end
        // of the allocation are silently dropped on translation failure.
        if ((s & 3) == 0) {
            __builtin_prefetch(cptr + 8 * stride, 0, 1);
        }

        // Leak-integrate: v += decay * (c - v)   (== v*(1-decay) + decay*c)
        v.x = fmaf(decay.x, c.x - v.x, v.x);
        v.y = fmaf(decay.y, c.y - v.y, v.y);
        v.z = fmaf(decay.z, c.z - v.z, v.z);
        v.w = fmaf(decay.w, c.w - v.w, v.w);

        // Threshold-fire. (v-1 > 0) <=> (v > 1) exactly in fp32 (Sterbenz).
        f32x4 spk;
        spk.x = (v.x > LIF_VTH) ? 1.0f : 0.0f;
        spk.y = (v.y > LIF_VTH) ? 1.0f : 0.0f;
        spk.z = (v.z > LIF_VTH) ? 1.0f : 0.0f;
        spk.w = (v.w > LIF_VTH) ? 1.0f : 0.0f;

        // Streaming store, non-temporal (written once, never re-read).
        __builtin_nontemporal_store(spk, (f32x4*)optr);

        // Zero-reset: v *= (1 - spike)
        v.x = (v.x > LIF_VTH) ? 0.0f : v.x;
        v.y = (v.y > LIF_VTH) ? 0.0f : v.y;
        v.z = (v.z > LIF_VTH) ? 0.0f : v.z;
        v.w = (v.w > LIF_VTH) ? 0.0f : v.w;

        cptr += stride;
        optr += stride;
    }
}

extern "C" void kernel_launch(void* const* d_in, const int* in_sizes, int n_in,
                              void* d_out, int out_size, void* d_ws, size_t ws_size,
                              hipStream_t stream) {
    const float* cur = (const float*)d_in[0];   // [T, B, N] f32
    const float* tau = (const float*)d_in[1];   // [N] f32
    float*       out = (float*)d_out;           // [T, B, N] f32

    const int N     = in_sizes[1];              // 2048 (power of two)
    const int total = in_sizes[0];              // T * B * N
    const int BN    = total / LIF_T;            // B * N = 262144

    const int threads = BN / 4;                 // one thread per 4 neurons
    const int block   = 256;                    // 8 wave32 waves per block
    const int grid    = (threads + block - 1) / block;

    lif_scan_kernel<<<grid, block, 0, stream>>>(cur, tau, out, BN, N - 1);
}